// BoxRegression_42125039239326
// MI455X (gfx1250) — compile-verified
//
#include <hip/hip_runtime.h>
#include <hip/hip_bf16.h>

// ---------------------------------------------------------------------------
// YOLO-layer loss for MI455X (gfx1250, wave32).
// Memory-bound streaming pass (B128-vectorized, 4 cells/iter) + sparse target
// corrections. Wave reductions ride V_WMMA_F32_16X16X4_F32 (ones-matrix
// column-sum trick), then ds_bpermute shuffles + LDS combine.
// ---------------------------------------------------------------------------

#define GDIM 256
#define NANCH 3
#define STRIDE_PX 8.0f          // 2048 / 256
#define GG 65536                // G*G
#define EPS16 1e-16f
#define EPS12 1e-12f

__device__ __constant__ float c_AW[NANCH] = {116.0f / 8.0f, 156.0f / 8.0f, 373.0f / 8.0f}; // 14.5, 19.5, 46.625
__device__ __constant__ float c_AH[NANCH] = { 90.0f / 8.0f, 198.0f / 8.0f, 326.0f / 8.0f}; // 11.25, 24.75, 40.75

typedef float v2f __attribute__((ext_vector_type(2)));
typedef float v8f __attribute__((ext_vector_type(8)));

__device__ __forceinline__ float sigmoidf_(float v) {
    return 1.0f / (1.0f + __expf(-v));
}

__device__ __forceinline__ float wave_reduce_add(float v) {
    v += __shfl_xor(v, 1, 32);
    v += __shfl_xor(v, 2, 32);
    v += __shfl_xor(v, 4, 32);
    v += __shfl_xor(v, 8, 32);
    v += __shfl_xor(v, 16, 32);
    return v;
}

// Sum one f32 value across the wave through the matrix unit:
// D = ones(16x4) x B(4x16): each column n of D = sum_k B[k,n], i.e. the WMMA
// folds lane n with lane n+16; the remaining 16 column-sums (replicated over
// rows) are collapsed with shuffles. EXEC must be all-ones at the call site.
__device__ __forceinline__ float wave_sum_wmma(float v) {
    v2f a; a[0] = 1.0f; a[1] = 1.0f;      // A = ones (16x4 f32, 2 VGPRs)
    v2f b; b[0] = v;    b[1] = 0.0f;      // B carries per-lane partials
    v8f c = {};
    v8f d = __builtin_amdgcn_wmma_f32_16x16x4_f32(
        /*neg_a=*/false, a, /*neg_b=*/false, b,
        /*c_mod=*/(short)0, c, /*reuse_a=*/false, /*reuse_b=*/false);
    // d[0]: lanes 0-15 hold colsums(0..15), lanes 16-31 hold the same values
    // (row 8 == row 0 since A is all-ones). Full-wave reduce => 2x total.
    return wave_reduce_add(d[0]) * 0.5f;
}

// ---------------------------------------------------------------------------
// Kernel 1: per-target preprocessing + obj scatter.
// ---------------------------------------------------------------------------
__global__ __launch_bounds__(256) void targets_kernel(
    const float* __restrict__ tg, int T,
    int* __restrict__ objmap, int* __restrict__ tcell, float* __restrict__ tf) {
    int t = blockIdx.x * blockDim.x + threadIdx.x;
    if (t >= T) return;
    float bF = tg[(size_t)t * 6 + 0];
    float gx = tg[(size_t)t * 6 + 2] * (float)GDIM;
    float gy = tg[(size_t)t * 6 + 3] * (float)GDIM;
    float gw = tg[(size_t)t * 6 + 4] * (float)GDIM;
    float gh = tg[(size_t)t * 6 + 5] * (float)GDIM;

    int best = 0; float bi = -1.0f;
#pragma unroll
    for (int a = 0; a < NANCH; ++a) {
        float inter = fminf(c_AW[a], gw) * fminf(c_AH[a], gh);
        float uni   = c_AW[a] * c_AH[a] + gw * gh - inter;
        float iou   = inter / (uni + EPS16);
        if (iou > bi) { bi = iou; best = a; }   // strict > keeps first max (argmax)
    }
    int b  = (int)bF;
    int gi = min(max((int)floorf(gx), 0), GDIM - 1);
    int gj = min(max((int)floorf(gy), 0), GDIM - 1);
    int cell = ((b * NANCH + best) * GDIM + gj) * GDIM + gi;

    atomicMax(&objmap[cell], t);   // deterministic winner among duplicates
    tcell[t] = cell;
    float* f = tf + (size_t)t * 8;
    f[0] = gx - (float)gi;                       // tx
    f[1] = gy - (float)gj;                       // ty
    f[2] = __logf(gw / c_AW[best] + EPS16);      // tw
    f[3] = __logf(gh / c_AH[best] + EPS16);      // th
    f[4] = gx; f[5] = gy; f[6] = gw; f[7] = gh;  // target box (grid units)
}

// ---------------------------------------------------------------------------
// Kernel 2: dense streaming pass, vectorized over groups of 4 cells.
// Per group: 5x global_load_b128 (one per channel) + 5x global_store_b128.
// Accumulates per block into blockSums:
//   [0] sum of bce(tconf=0) over ALL cells
//   [1] sum of pred_conf over ALL cells
//   [2] count of pred_conf > 0.5 over ALL cells
// ---------------------------------------------------------------------------
__global__ __launch_bounds__(256) void dense_kernel(
    const float* __restrict__ x, float* __restrict__ out,
    float* __restrict__ blockSums, int Ngroups) {
    float s_bce = 0.0f, s_conf = 0.0f, s_c50 = 0.0f;
    int stride = gridDim.x * blockDim.x;
    for (int g = blockIdx.x * blockDim.x + threadIdx.x; g < Ngroups; g += stride) {
        int i0    = g << 2;                  // first cell of the 4-cell group
        int plane = i0 >> 16;                // b*3 + a  (group never crosses a plane)
        int low   = i0 & (GG - 1);
        int a     = plane % NANCH;
        float awS = c_AW[a] * STRIDE_PX;
        float ahS = c_AH[a] * STRIDE_PX;
        size_t xb = (size_t)plane * (5 * GG) + (size_t)low;   // 16B-aligned (low % 4 == 0)
        float4 vtx = *(const float4*)(x + xb);
        float4 vty = *(const float4*)(x + xb + 1 * GG);
        float4 vw  = *(const float4*)(x + xb + 2 * GG);
        float4 vh  = *(const float4*)(x + xb + 3 * GG);
        float4 vc  = *(const float4*)(x + xb + 4 * GG);
        __builtin_prefetch(x + xb + (size_t)stride * 4, 0, 0);  // global_prefetch_b8

        const float* ptx = (const float*)&vtx;
        const float* pty = (const float*)&vty;
        const float* pw  = (const float*)&vw;
        const float* ph  = (const float*)&vh;
        const float* pc  = (const float*)&vc;
        float gyf = (float)(low >> 8);
        float gx0 = (float)(low & (GDIM - 1));   // row groups: 256 % 4 == 0

        float ox[20];
#pragma unroll
        for (int k = 0; k < 4; ++k) {
            float cx   = sigmoidf_(ptx[k]);
            float cy   = sigmoidf_(pty[k]);
            float conf = sigmoidf_(pc[k]);
            ox[k * 5 + 0] = (gx0 + (float)k + cx) * STRIDE_PX;
            ox[k * 5 + 1] = (gyf + cy) * STRIDE_PX;
            ox[k * 5 + 2] = __expf(pw[k]) * awS;
            ox[k * 5 + 3] = __expf(ph[k]) * ahS;
            ox[k * 5 + 4] = conf;
            s_bce  += -__logf(1.0f - conf + EPS12);
            s_conf += conf;
            s_c50  += (conf > 0.5f) ? 1.0f : 0.0f;
        }
        // 20 contiguous floats at byte offset g*80: 16B-aligned -> 5x b128 store
        float4* op = (float4*)(out + (size_t)g * 20);
#pragma unroll
        for (int k = 0; k < 5; ++k)
            op[k] = make_float4(ox[k * 4 + 0], ox[k * 4 + 1], ox[k * 4 + 2], ox[k * 4 + 3]);
    }

    __shared__ float lacc[3];
    if (threadIdx.x < 3) lacc[threadIdx.x] = 0.0f;
    __syncthreads();

    // Uniform control flow here => EXEC all-ones, as WMMA requires.
    float r0 = wave_sum_wmma(s_bce);
    float r1 = wave_sum_wmma(s_conf);
    float r2 = wave_sum_wmma(s_c50);
    if ((threadIdx.x & 31) == 0) {
        atomicAdd(&lacc[0], r0);
        atomicAdd(&lacc[1], r1);
        atomicAdd(&lacc[2], r2);
    }
    __syncthreads();
    if (threadIdx.x < 3) blockSums[(size_t)blockIdx.x * 3 + threadIdx.x] = lacc[threadIdx.x];
}

// ---------------------------------------------------------------------------
// Kernel 3: sparse corrections, one thread per (target, anchor).
// acc layout: 0 rm_bce, 1 rm_conf, 2 rm_cnt, 3 o_bce, 4 o_conf, 5 o_cnt,
//             6 lx, 7 ly, 8 lw, 9 lh, 10 d50, 11 d75
// ---------------------------------------------------------------------------
__global__ __launch_bounds__(256) void corr_kernel(
    const float* __restrict__ x, const int* __restrict__ tcell,
    const float* __restrict__ tf, const int* __restrict__ objmap,
    unsigned* __restrict__ rmbits, float* __restrict__ acc, int T) {
    int idx = blockIdx.x * blockDim.x + threadIdx.x;
    float p[12];
#pragma unroll
    for (int q = 0; q < 12; ++q) p[q] = 0.0f;

    if (idx < T * NANCH) {
        int t = idx / NANCH;
        int a = idx % NANCH;
        int cell  = tcell[t];
        int plane = cell >> 16;
        int best  = plane % NANCH;
        const float* f = tf + (size_t)t * 8;
        float gw = f[6], gh = f[7];

        float inter = fminf(c_AW[a], gw) * fminf(c_AH[a], gh);
        float uni   = c_AW[a] * c_AH[a] + gw * gh - inter;
        float iou_a = inter / (uni + EPS16);

        int  cell_a     = cell + (a - best) * GG;
        bool isObjAnch  = (a == best);
        bool qualRemove = isObjAnch || (iou_a > 0.5f);   // IGNORE_THRES

        if (qualRemove) {
            unsigned word = (unsigned)cell_a >> 5;
            unsigned bit  = 1u << (cell_a & 31);
            unsigned old  = atomicOr(&rmbits[word], bit);
            if (!(old & bit)) {   // exactly-once removal from the no-obj set
                size_t xb = (size_t)(cell_a >> 16) * (5 * GG) + (size_t)(cell_a & (GG - 1));
                float conf = sigmoidf_(x[xb + 4 * GG]);
                p[0] += -__logf(1.0f - conf + EPS12);
                p[1] += conf;
                p[2] += 1.0f;
            }
        }

        if (isObjAnch && objmap[cell_a] == t) {          // scatter winner
            size_t xb = (size_t)plane * (5 * GG) + (size_t)(cell_a & (GG - 1));
            float cx   = sigmoidf_(x[xb]);
            float cy   = sigmoidf_(x[xb + 1 * GG]);
            float w    = x[xb + 2 * GG];
            float h    = x[xb + 3 * GG];
            float conf = sigmoidf_(x[xb + 4 * GG]);
            p[3] += -__logf(conf + EPS12);
            p[4] += conf;
            p[5] += 1.0f;
            float dx = cx - f[0], dy = cy - f[1], dw = w - f[2], dh = h - f[3];
            p[6] += dx * dx; p[7] += dy * dy; p[8] += dw * dw; p[9] += dh * dh;
            // IoU(pred box, target box) in grid units
            float gi = (float)(cell_a & (GDIM - 1));
            float gj = (float)((cell_a >> 8) & (GDIM - 1));
            float px = gi + cx, py = gj + cy;
            float pwv = __expf(w) * c_AW[a], phv = __expf(h) * c_AH[a];
            float b1x1 = px - pwv * 0.5f, b1x2 = px + pwv * 0.5f;
            float b1y1 = py - phv * 0.5f, b1y2 = py + phv * 0.5f;
            float b2x1 = f[4] - gw * 0.5f, b2x2 = f[4] + gw * 0.5f;
            float b2y1 = f[5] - gh * 0.5f, b2y2 = f[5] + gh * 0.5f;
            float iw = fmaxf(fminf(b1x2, b2x2) - fmaxf(b1x1, b2x1), 0.0f);
            float ih = fmaxf(fminf(b1y2, b2y2) - fmaxf(b1y1, b2y1), 0.0f);
            float in2 = iw * ih;
            float a1 = (b1x2 - b1x1) * (b1y2 - b1y1);
            float a2 = (b2x2 - b2x1) * (b2y2 - b2y1);
            float iou_s = in2 / (a1 + a2 - in2 + EPS16);
            float det = (conf > 0.5f) ? 1.0f : 0.0f;
            p[10] += (iou_s > 0.5f)  ? det : 0.0f;
            p[11] += (iou_s > 0.75f) ? det : 0.0f;
        }
    }

#pragma unroll
    for (int q = 0; q < 12; ++q) p[q] = wave_reduce_add(p[q]);
    if ((threadIdx.x & 31) == 0) {
#pragma unroll
        for (int q = 0; q < 12; ++q) atomicAdd(&acc[q], p[q]);
    }
}

// ---------------------------------------------------------------------------
// Kernel 4: fold blockSums + combine with sparse acc, emit loss + metrics.
// ---------------------------------------------------------------------------
__global__ __launch_bounds__(256) void final_kernel(
    const float* __restrict__ blockSums, int nblocks,
    const float* __restrict__ acc, float* __restrict__ outScal, float Ncells) {
    float r0 = 0.0f, r1 = 0.0f, r2 = 0.0f;
    for (int i = threadIdx.x; i < nblocks; i += blockDim.x) {
        r0 += blockSums[(size_t)i * 3 + 0];
        r1 += blockSums[(size_t)i * 3 + 1];
        r2 += blockSums[(size_t)i * 3 + 2];
    }
    __shared__ float sw[8][3];
    r0 = wave_reduce_add(r0);
    r1 = wave_reduce_add(r1);
    r2 = wave_reduce_add(r2);
    int wid = threadIdx.x >> 5;
    if ((threadIdx.x & 31) == 0) { sw[wid][0] = r0; sw[wid][1] = r1; sw[wid][2] = r2; }
    __syncthreads();
    if (threadIdx.x == 0) {
        float S_bce = 0.0f, S_conf = 0.0f, S_c50 = 0.0f;
        for (int wv = 0; wv < 8; ++wv) { S_bce += sw[wv][0]; S_conf += sw[wv][1]; S_c50 += sw[wv][2]; }
        float rm_bce = acc[0], rm_conf = acc[1], rm_cnt = acc[2];
        float o_bce = acc[3], o_conf = acc[4], o_cnt = acc[5];
        float lx = acc[6], ly = acc[7], lw = acc[8], lh = acc[9];
        float d50 = acc[10], d75 = acc[11];

        float dobj = fmaxf(o_cnt, 1.0f);
        float n_noobj = Ncells - rm_cnt;
        float dnob = fmaxf(n_noobj, 1.0f);
        float loss_x = lx / dobj, loss_y = ly / dobj;
        float loss_w = lw / dobj, loss_h = lh / dobj;
        float loss_bbox = loss_x + loss_y + loss_w + loss_h;
        float loss_conf_obj   = o_bce / dobj;
        float loss_conf_noobj = (S_bce - rm_bce) / dnob;
        float loss_conf = 100.0f * loss_conf_obj + 1.0f * loss_conf_noobj;
        float loss_layer = loss_bbox + loss_conf;
        float conf_obj   = o_conf / dobj;
        float conf_noobj = (S_conf - rm_conf) / dnob;
        float precision = d50 / (S_c50 + EPS16);
        float recall50  = d50 / (o_cnt + EPS16);
        float recall75  = d75 / (o_cnt + EPS16);

        outScal[0]  = loss_layer;     // scalar return value
        outScal[1]  = loss_x;         // metrics[0..11]
        outScal[2]  = loss_y;
        outScal[3]  = loss_w;
        outScal[4]  = loss_h;
        outScal[5]  = loss_bbox;
        outScal[6]  = loss_conf;
        outScal[7]  = loss_layer;
        outScal[8]  = conf_obj;
        outScal[9]  = conf_noobj;
        outScal[10] = precision;
        outScal[11] = recall50;
        outScal[12] = recall75;
    }
}

// ---------------------------------------------------------------------------
extern "C" void kernel_launch(void* const* d_in, const int* in_sizes, int n_in,
                              void* d_out, int out_size, void* d_ws, size_t ws_size,
                              hipStream_t stream) {
    (void)n_in; (void)out_size; (void)ws_size;
    const float* x       = (const float*)d_in[0];
    const float* targets = (const float*)d_in[1];
    float* out = (float*)d_out;

    int T = in_sizes[1] / 6;                           // 4096
    int B = in_sizes[0] / (NANCH * 5 * GDIM * GDIM);   // 32
    int N = B * NANCH * GDIM * GDIM;                   // 6,291,456 cells
    int Ngroups = N >> 2;                              // 4 cells per group

    const int GROUPS_PER_THREAD = 2;                   // 8 cells per thread
    int nblocks = (Ngroups + 256 * GROUPS_PER_THREAD - 1) / (256 * GROUPS_PER_THREAD);

    // Workspace carve-up
    char* w = (char*)d_ws;
    size_t o = 0;
    float*    acc       = (float*)(w + o);    o += 256;
    int*      tcell     = (int*)(w + o);      o += (((size_t)T * 4) + 255) & ~(size_t)255;
    float*    tf        = (float*)(w + o);    o += (((size_t)T * 32) + 255) & ~(size_t)255;
    int*      objmap    = (int*)(w + o);      o += (size_t)N * 4;
    unsigned* rmbits    = (unsigned*)(w + o); o += (size_t)(N / 32) * 4;
    float*    blockSums = (float*)(w + o);    o += (size_t)nblocks * 3 * 4;

    // Per-call deterministic init (graph-capture safe stream ops)
    hipMemsetAsync(acc, 0, 256, stream);
    hipMemsetAsync(objmap, 0xFF, (size_t)N * 4, stream);        // -1
    hipMemsetAsync(rmbits, 0, (size_t)(N / 32) * 4, stream);

    targets_kernel<<<(T + 255) / 256, 256, 0, stream>>>(targets, T, objmap, tcell, tf);
    dense_kernel<<<nblocks, 256, 0, stream>>>(x, out, blockSums, Ngroups);
    corr_kernel<<<(T * NANCH + 255) / 256, 256, 0, stream>>>(x, tcell, tf, objmap, rmbits, acc, T);
    final_kernel<<<1, 256, 0, stream>>>(blockSums, nblocks, acc, out + (size_t)N * 5, (float)N);
}